// LogLikelihoodLossWithCensoring_27797028339926
// MI455X (gfx1250) — compile-verified
//
#include <hip/hip_runtime.h>

typedef __attribute__((ext_vector_type(2))) float v2f;
typedef __attribute__((ext_vector_type(8))) float v8f;
typedef __attribute__((ext_vector_type(4))) float f4;
typedef __attribute__((ext_vector_type(4))) int   i4;

// ---------------------------------------------------------------------------
// Kernel 1: build fused coefficient table in workspace, zero the accumulator.
// Entry [risk*256 + t] = (A, B, C, 0) such that
//     contribution = A*log(lam) + B*lam + C
//   risk==1 : A = t,           B = -1,   C = -gammaln(t+1)
//   risk==0 : A = -t(t+1)/2,   B = t+1,  C = (t+1) + cumsum_{j<=t} gammaln(j+1)
// ---------------------------------------------------------------------------
__global__ __launch_bounds__(256) void pois_init(f4* __restrict__ tabs,
                                                 float* __restrict__ out) {
  __shared__ float sc[256];
  const int tid = threadIdx.x;
  const float t = (float)tid;
  const float g = lgammaf(t + 1.0f);            // gammaln(t+1)
  sc[tid] = g;
  __syncthreads();
  // Hillis-Steele inclusive scan -> cumsum_{j=0..t} gammaln(j+1)
  for (int off = 1; off < 256; off <<= 1) {
    float v = (tid >= off) ? sc[tid - off] : 0.0f;
    __syncthreads();
    sc[tid] += v;
    __syncthreads();
  }
  const float cum = sc[tid];

  f4 e0;                                        // risk == 0 (censored / surv)
  e0.x = -t * (t + 1.0f) * 0.5f;
  e0.y = t + 1.0f;
  e0.z = (t + 1.0f) + cum;
  e0.w = 0.0f;
  tabs[tid] = e0;

  f4 e1;                                        // risk == 1 (observed / loglik)
  e1.x = t;
  e1.y = -1.0f;
  e1.z = -g;
  e1.w = 0.0f;
  tabs[256 + tid] = e1;

  if (tid == 0) out[0] = 0.0f;
}

// ---------------------------------------------------------------------------
// Kernel 2: streaming reduction. 128-bit non-temporal loads, one fused
// ds_load_b128 table lookup per element, WMMA ones-matrix wave reduction,
// one atomicAdd per block.
// ---------------------------------------------------------------------------
__global__ __launch_bounds__(256) void pois_main(const float* __restrict__ lam_in,
                                                 const int* __restrict__ tgt,
                                                 const int* __restrict__ rsk,
                                                 const f4* __restrict__ tabs,
                                                 float* __restrict__ out,
                                                 int nv, float scale) {
  __shared__ f4 s_tab[512];                     // 8 KB fused coefficient table
  __shared__ float s_wave[8];
  const int tid = threadIdx.x;
  s_tab[tid]       = tabs[tid];
  s_tab[tid + 256] = tabs[tid + 256];
  __syncthreads();

  const f4* lam4 = (const f4*)lam_in;
  const i4* t4   = (const i4*)tgt;
  const i4* r4   = (const i4*)rsk;

  float acc = 0.0f;
  const int stride = gridDim.x * blockDim.x;
  for (int i = blockIdx.x * blockDim.x + tid; i < nv; i += stride) {
    f4 l  = __builtin_nontemporal_load(&lam4[i]);
    i4 tt = __builtin_nontemporal_load(&t4[i]);
    i4 rr = __builtin_nontemporal_load(&r4[i]);

    {
      f4 e = s_tab[(rr.x << 8) | tt.x];
      acc += __builtin_fmaf(e.x, __logf(l.x), __builtin_fmaf(e.y, l.x, e.z));
    }
    {
      f4 e = s_tab[(rr.y << 8) | tt.y];
      acc += __builtin_fmaf(e.x, __logf(l.y), __builtin_fmaf(e.y, l.y, e.z));
    }
    {
      f4 e = s_tab[(rr.z << 8) | tt.z];
      acc += __builtin_fmaf(e.x, __logf(l.z), __builtin_fmaf(e.y, l.z, e.z));
    }
    {
      f4 e = s_tab[(rr.w << 8) | tt.w];
      acc += __builtin_fmaf(e.x, __logf(l.w), __builtin_fmaf(e.y, l.w, e.z));
    }
  }

  // Wave-level reduction via the matrix pipe:
  // A (16x4) holds the wave's 32 partials (other 32 slots zero), B (4x16) = 1s.
  // D[m][n] = rowsum_A[m] for every n; each lane's 8 D regs hold 8 distinct
  // row sums, so local sum + the opposite half-wave's value (xor 16) is the
  // full 32-lane sum.
  v2f a; a.x = acc;  a.y = 0.0f;
  v2f b; b.x = 1.0f; b.y = 1.0f;
  v8f c = {};
  v8f d = __builtin_amdgcn_wmma_f32_16x16x4_f32(
      /*neg_a=*/false, a, /*neg_b=*/false, b,
      /*c_mod=*/(short)0, c, /*reuse_a=*/false, /*reuse_b=*/false);
  float s = 0.0f;
#pragma unroll
  for (int k = 0; k < 8; ++k) s += d[k];
  const float wave_total = s + __shfl_xor(s, 16, 32);

  const int lane = tid & 31;
  const int wv   = tid >> 5;
  if (lane == 0) s_wave[wv] = wave_total;
  __syncthreads();
  if (tid == 0) {
    float bsum = 0.0f;
#pragma unroll
    for (int w = 0; w < 8; ++w) bsum += s_wave[w];
    atomicAdd(out, bsum * scale);
  }
}

// ---------------------------------------------------------------------------
extern "C" void kernel_launch(void* const* d_in, const int* in_sizes, int n_in,
                              void* d_out, int out_size, void* d_ws, size_t ws_size,
                              hipStream_t stream) {
  const float* lam = (const float*)d_in[0];   // inputs (N,1) f32
  const int*   tgt = (const int*)d_in[1];     // targets (N,) i32
  const int*   rsk = (const int*)d_in[2];     // risk (N,) i32
  float* out  = (float*)d_out;
  f4*    tabs = (f4*)d_ws;                    // 512 float4 (8 KB) of scratch

  const int n  = in_sizes[0];                 // N = 8388608
  const int nv = n >> 2;                      // float4 groups

  pois_init<<<1, 256, 0, stream>>>(tabs, out);

  int grid = 4096;                            // 32768 wave32s -> saturate HBM
  if (grid > (nv + 255) / 256) grid = (nv + 255) / 256;
  pois_main<<<grid, 256, 0, stream>>>(lam, tgt, rsk, tabs, out, nv,
                                      -1.0f / (float)n);
}